// HSL1Loss_1022202216838
// MI455X (gfx1250) — compile-verified
//
#include <hip/hip_runtime.h>

#define NUM_LABELS 64
#define B_SZ 16
#define HW (1024 * 1024)
#define BLOCKS_PER_BATCH 128
#define CHUNK (HW / BLOCKS_PER_BATCH) /* 8192 pixels per block */
#define THREADS 256
#define NSEG (B_SZ * NUM_LABELS) /* 1024 segments */

typedef __attribute__((ext_vector_type(2))) float v2f;
typedef __attribute__((ext_vector_type(8))) float v8f;

// ---------------------------------------------------------------------------
// Kernel 0: zero the workspace accumulators (harness poisons d_ws with 0xAA).
// ---------------------------------------------------------------------------
__global__ void hsl_zero(float* __restrict__ g_sum, unsigned* __restrict__ g_cnt) {
    int i = blockIdx.x * blockDim.x + threadIdx.x;
    if (i < NSEG) {
        g_sum[i] = 0.0f;
        g_cnt[i] = 0u;
    }
}

// ---------------------------------------------------------------------------
// Kernel 1: streaming histogram. 128 MiB read @ 23.3 TB/s => ~5.8 us bound.
// 64 bins == 64 LDS banks; per-pixel ds_add_f32 / ds_add_u32 (no-return),
// one global atomic per (block, bin) flush at the end.
// ---------------------------------------------------------------------------
__device__ __forceinline__ void hsl_acc(float f, int g,
                                        float* s_sum, unsigned* s_cnt) {
    int l = (g < 255) ? g : 0;   // IGNORE_LABEL pixels -> segment 0 (dropped later)
    l &= (NUM_LABELS - 1);       // safety clamp; labels are in [0,64)
    float d = f - (float)l;
    atomicAdd(&s_sum[l], d * d); // ds_add_f32 (no return)
    atomicAdd(&s_cnt[l], 1u);    // ds_add_u32 (no return)
}

__global__ __launch_bounds__(THREADS) void hsl_phase1(
        const float* __restrict__ feat, const int* __restrict__ gt,
        float* __restrict__ g_sum, unsigned* __restrict__ g_cnt) {
    __shared__ float    s_sum[NUM_LABELS];
    __shared__ unsigned s_cnt[NUM_LABELS];

    const int t = threadIdx.x;
    if (t < NUM_LABELS) { s_sum[t] = 0.0f; s_cnt[t] = 0u; }
    __syncthreads();

    const int b   = blockIdx.x / BLOCKS_PER_BATCH;
    const int blk = blockIdx.x % BLOCKS_PER_BATCH;
    const size_t base = (size_t)b * HW + (size_t)blk * CHUNK;

    const float4* __restrict__ f4 = (const float4*)(feat + base);
    const int4*   __restrict__ g4 = (const int4*)(gt + base);
    const int nVec = CHUNK / 4; // 2048 float4/int4 vectors per block

    for (int i = t; i < nVec; i += THREADS) {
        __builtin_prefetch(f4 + i + THREADS, 0, 0); // global_prefetch_b8
        __builtin_prefetch(g4 + i + THREADS, 0, 0);
        float4 f = f4[i];   // global_load_b128
        int4   g = g4[i];   // global_load_b128
        hsl_acc(f.x, g.x, s_sum, s_cnt);
        hsl_acc(f.y, g.y, s_sum, s_cnt);
        hsl_acc(f.z, g.z, s_sum, s_cnt);
        hsl_acc(f.w, g.w, s_sum, s_cnt);
    }
    __syncthreads();

    if (t < NUM_LABELS) {
        unsafeAtomicAdd(&g_sum[b * NUM_LABELS + t], s_sum[t]); // global_atomic_add_f32
        atomicAdd(&g_cnt[b * NUM_LABELS + t], s_cnt[t]);       // global_atomic_add_u32
    }
}

// ---------------------------------------------------------------------------
// Kernel 2: finalize with WMMA. One wave32. The 1024 per-instance means are
// summed via 16 accumulating V_WMMA_F32_16X16X4_F32 ops:
//   A_i = 16x4 chunk of x[], B = ones(4x16)  =>  D[m][n] = sum_k,i A_i[m][k].
// A layout (32-bit A 16x4): lane L(0..15): VGPR0=K0, VGPR1=K1 for M=L;
//                           lane L(16..31): VGPR0=K2, VGPR1=K3 for M=L-16.
// D layout (32-bit C/D 16x16): VGPR r: lane0 -> (M=r,N=0), lane16 -> (M=r+8,N=0).
// ---------------------------------------------------------------------------
__device__ __forceinline__ float hsl_val(const float* g_sum,
                                         const unsigned* g_cnt, int s) {
    int l = s & (NUM_LABELS - 1);        // chunk i == batch i, offset == label
    float    sm = g_sum[s];
    unsigned cv = g_cnt[s];
    float v = (l != 0 && cv > 0u) ? (sm / (float)cv) : 0.0f;
    return v;
}

__global__ void hsl_phase2(const float* __restrict__ g_sum,
                           const unsigned* __restrict__ g_cnt,
                           float* __restrict__ out) {
    const int lane = threadIdx.x;        // 0..31, EXEC all ones (no divergence)
    const int m  = lane & 15;
    const int kh = lane >> 4;            // 0 -> K{0,1}, 1 -> K{2,3}

    v2f bmat;
    bmat.x = 1.0f;
    bmat.y = 1.0f;
    v8f c = {};

    #pragma unroll
    for (int i = 0; i < B_SZ; ++i) {     // 16 chunks of 64 values
        const int s0 = i * NUM_LABELS + m * 4 + 2 * kh;
        v2f a;
        a.x = hsl_val(g_sum, g_cnt, s0);
        a.y = hsl_val(g_sum, g_cnt, s0 + 1);
        c = __builtin_amdgcn_wmma_f32_16x16x4_f32(
                /*neg_a=*/false, a, /*neg_b=*/false, bmat,
                /*c_mod=*/(short)0, c, /*reuse_a=*/false, /*reuse_b=*/false);
    }

    // Sum column N=0 of D over M=0..15.
    float total = 0.0f;
    #pragma unroll
    for (int r = 0; r < 8; ++r) {
        total += __shfl(c[r], 0, 32);    // (M=r,   N=0)
        total += __shfl(c[r], 16, 32);   // (M=r+8, N=0)
    }
    if (lane == 0) out[0] = total * (1.0f / (float)B_SZ);
}

// ---------------------------------------------------------------------------
extern "C" void kernel_launch(void* const* d_in, const int* in_sizes, int n_in,
                              void* d_out, int out_size, void* d_ws, size_t ws_size,
                              hipStream_t stream) {
    const float* feat = (const float*)d_in[0];
    const int*   gt   = (const int*)d_in[1];

    float*    g_sum = (float*)d_ws;
    unsigned* g_cnt = (unsigned*)((char*)d_ws + NSEG * sizeof(float));

    hsl_zero<<<(NSEG + 255) / 256, 256, 0, stream>>>(g_sum, g_cnt);
    hsl_phase1<<<B_SZ * BLOCKS_PER_BATCH, THREADS, 0, stream>>>(feat, gt, g_sum, g_cnt);
    hsl_phase2<<<1, 32, 0, stream>>>(g_sum, g_cnt, (float*)d_out);
}